// GCN_55765855371408
// MI455X (gfx1250) — compile-verified
//
#include <hip/hip_runtime.h>
#include <hip/hip_bf16.h>

typedef __attribute__((ext_vector_type(2))) float v2f;
typedef __attribute__((ext_vector_type(8))) float v8f;

#define N_NODES 50000
#define N_EDGES 800000
#define D_HID   128
#define D_OUT   64

// ---------------------------------------------------------------------------
// Degree count: deg[dst[e]] += 1
// ---------------------------------------------------------------------------
__global__ void degree_kernel(const int* __restrict__ dst, float* __restrict__ deg, int E) {
    int e = blockIdx.x * blockDim.x + threadIdx.x;
    if (e < E) atomicAdd(&deg[dst[e]], 1.0f);
}

// dinv[i] = rsqrt(deg[i] + 1)   (self loop contributes 1; always > 0)
__global__ void dinv_kernel(float* __restrict__ deg, int n) {
    int i = blockIdx.x * blockDim.x + threadIdx.x;
    if (i < n) deg[i] = rsqrtf(deg[i] + 1.0f);
}

// ---------------------------------------------------------------------------
// WMMA f32 GEMM: H[M,N] = X[M,K=128] @ W[128,N] (+ bias if non-null)
// One wave per 16x16 output tile. K = 128 fixed (32 WMMA f32 16x16x4 steps).
// Layouts per CDNA5 ISA 7.12.2:
//   A 16x4 f32: lane(l&15)=row M; (l>>4) selects K pair {0,1} vs {2,3}
//   B 4x16 f32: lane(l&15)=col N; (l>>4) selects K pair
//   D 16x16 f32: VGPR v holds row v + 8*(l>>4), col = l&15
// ---------------------------------------------------------------------------
template <int N>
__global__ void gemm_wmma_f32(const float* __restrict__ X,
                              const float* __restrict__ W,
                              const float* __restrict__ bias,
                              float* __restrict__ H, int M) {
    const int K = 128;
    int wave  = (blockIdx.x * blockDim.x + threadIdx.x) >> 5;
    int lane  = threadIdx.x & 31;
    const int ntiles = N >> 4;
    int tileM = wave / ntiles;
    int tileN = wave - tileM * ntiles;
    if (tileM * 16 >= M) return;              // wave-uniform exit (EXEC all-ones otherwise)

    int sub = lane & 15;
    int kh  = lane >> 4;                      // 0 or 1: which K-pair of the 4
    int row = tileM * 16 + sub;               // A row this lane feeds
    int col = tileN * 16 + sub;               // B col this lane feeds

    const float* __restrict__ xrow = X + (size_t)row * K + 2 * kh;
    const float* __restrict__ wcol = W + (size_t)(2 * kh) * N + col;

    v8f acc = {};
#pragma unroll
    for (int k = 0; k < K; k += 4) {
        v2f a, b;
        a.x = xrow[k];                        // X[row][k + 2*kh]
        a.y = xrow[k + 1];                    // X[row][k + 2*kh + 1]
        b.x = wcol[(size_t)k * N];            // W[k + 2*kh][col]
        b.y = wcol[(size_t)(k + 1) * N];      // W[k + 2*kh + 1][col]
        acc = __builtin_amdgcn_wmma_f32_16x16x4_f32(
            /*neg_a=*/false, a, /*neg_b=*/false, b,
            /*c_mod=*/(short)0, acc, /*reuse_a=*/false, /*reuse_b=*/false);
    }

    float badd = bias ? bias[col] : 0.0f;
    int orow = tileM * 16 + kh * 8;
#pragma unroll
    for (int v = 0; v < 8; ++v)
        H[(size_t)(orow + v) * N + col] = acc[v] + badd;
}

// ---------------------------------------------------------------------------
// Edge scatter: acc[dst] += h[src] * dinv[src]*dinv[dst]   (one wave per edge)
// ---------------------------------------------------------------------------
__global__ void scatter_kernel(const float* __restrict__ H,
                               const int* __restrict__ src,
                               const int* __restrict__ dst,
                               const float* __restrict__ dinv,
                               float* __restrict__ acc, int E) {
    int wave = (blockIdx.x * blockDim.x + threadIdx.x) >> 5;
    int lane = threadIdx.x & 31;
    if (wave >= E) return;
    int s = src[wave];
    int d = dst[wave];
    float nrm = dinv[s] * dinv[d];
    const float* __restrict__ hs = H + (size_t)s * D_HID;
    float* __restrict__ ad = acc + (size_t)d * D_HID;
#pragma unroll
    for (int j = 0; j < 4; ++j) {
        int f = lane + 32 * j;
        atomicAdd(&ad[f], hs[f] * nrm);
    }
}

// ---------------------------------------------------------------------------
// Epilogue: out = relu(acc + h*dinv^2 (self loop) + bias)
// ---------------------------------------------------------------------------
__global__ void epilogue_kernel(const float* __restrict__ acc,
                                const float* __restrict__ H,
                                const float* __restrict__ dinv,
                                const float* __restrict__ b,
                                float* __restrict__ out, int n_nodes) {
    long long idx = (long long)blockIdx.x * blockDim.x + threadIdx.x;
    long long total = (long long)n_nodes * D_HID;
    if (idx >= total) return;
    int node = (int)(idx >> 7);
    int f    = (int)(idx & (D_HID - 1));
    float di = dinv[node];
    float v  = acc[idx] + H[idx] * di * di + b[f];
    out[idx] = v > 0.0f ? v : 0.0f;
}

// ---------------------------------------------------------------------------
extern "C" void kernel_launch(void* const* d_in, const int* in_sizes, int n_in,
                              void* d_out, int out_size, void* d_ws, size_t ws_size,
                              hipStream_t stream) {
    const float* x  = (const float*)d_in[0];   // [50000,128]
    const int*   ei = (const int*)d_in[1];     // [2,800000]
    const float* Ws = (const float*)d_in[2];   // [3,128,128]
    const float* bs = (const float*)d_in[3];   // [3,128]
    const float* Wl = (const float*)d_in[4];   // [128,64]
    const float* bl = (const float*)d_in[5];   // [64]

    const int N = N_NODES, E = N_EDGES;
    const int* src = ei;
    const int* dst = ei + E;

    // workspace layout (floats)
    float* ws   = (float*)d_ws;
    float* dinv = ws;                               // 50176 (padded)
    float* cur  = dinv + 50176;                     // N*128
    float* h    = cur  + (size_t)N * D_HID;         // N*128
    float* accb = h    + (size_t)N * D_HID;         // N*128

    // 1) normalization coefficients
    hipMemsetAsync(dinv, 0, (size_t)N * sizeof(float), stream);
    degree_kernel<<<(E + 255) / 256, 256, 0, stream>>>(dst, dinv, E);
    dinv_kernel<<<(N + 255) / 256, 256, 0, stream>>>(dinv, N);

    // 2) three GCN layers
    const float* in = x;
    for (int l = 0; l < 3; ++l) {
        int waves = (N / 16) * (D_HID / 16);
        int thr   = waves * 32;
        gemm_wmma_f32<D_HID><<<(thr + 255) / 256, 256, 0, stream>>>(
            in, Ws + (size_t)l * D_HID * D_HID, nullptr, h, N);

        hipMemsetAsync(accb, 0, (size_t)N * D_HID * sizeof(float), stream);

        long long sthr = (long long)E * 32;
        scatter_kernel<<<(int)((sthr + 255) / 256), 256, 0, stream>>>(
            h, src, dst, dinv, accb, E);

        long long ethr = (long long)N * D_HID;
        epilogue_kernel<<<(int)((ethr + 255) / 256), 256, 0, stream>>>(
            accb, h, dinv, bs + (size_t)l * D_HID, cur, N);
        in = cur;
    }

    // 3) final linear with fused bias -> d_out (f32 [50000,64])
    int waves2 = (N / 16) * (D_OUT / 16);
    int thr2   = waves2 * 32;
    gemm_wmma_f32<D_OUT><<<(thr2 + 255) / 256, 256, 0, stream>>>(
        cur, Wl, bl, (float*)d_out, N);
}